// DisentangledSelfAttention_18373870092786
// MI455X (gfx1250) — compile-verified
//
#include <hip/hip_runtime.h>
#include <hip/hip_bf16.h>

typedef float v2f __attribute__((ext_vector_type(2)));
typedef float v8f __attribute__((ext_vector_type(8)));

#define S_ 1024
#define H_ 12
#define D_ 64
#define HID_ 768
#define TK_ 512   // 2*K
#define B_ 4

static __device__ __forceinline__ int iclamp(int x, int lo, int hi) {
    return x < lo ? lo : (x > hi ? hi : x);
}

// Generic C[m,n] = sum_k A[m,k]*B[n,k] (+bias[n]), fp32 WMMA 16x16x4.
// z = batch index; b = z/Hdiv, h = z%Hdiv;  A += b*sAb + h*sAh; B += h*sBh; C += z*sCz.
// Each wave computes one 16x16 tile; block = 4 waves covering 64 output columns.
__global__ void wmma_gemm_abt(const float* __restrict__ A, const float* __restrict__ Bm,
                              const float* __restrict__ bias, float* __restrict__ C,
                              int M, int N, int Kd, int lda, int ldb, int ldc,
                              int Hdiv, long long sAb, long long sAh,
                              long long sBh, long long sCz)
{
    const int wave = threadIdx.x >> 5;
    const int lane = threadIdx.x & 31;
    const int half = lane >> 4;
    const int l    = lane & 15;
    const int tn = (blockIdx.x * 4 + wave) * 16;
    const int tm = blockIdx.y * 16;
    const int z  = blockIdx.z;
    if (tn >= N || tm >= M) return;

    const int bb = z / Hdiv, hh = z % Hdiv;
    const float* Ap = A + (long long)bb * sAb + (long long)hh * sAh + (long long)(tm + l) * lda;
    const float* Bp = Bm + (long long)hh * sBh + (long long)(tn + l) * ldb;
    float* Cp = C + (long long)z * sCz;

    v8f acc = {};
    for (int kk = 0; kk < Kd; kk += 4) {
        const int k0 = kk + half * 2;
        v2f a; a.x = Ap[k0]; a.y = Ap[k0 + 1];
        v2f b; b.x = Bp[k0]; b.y = Bp[k0 + 1];
        acc = __builtin_amdgcn_wmma_f32_16x16x4_f32(false, a, false, b,
                                                    (short)0, acc, false, false);
    }
#pragma unroll
    for (int r = 0; r < 8; ++r) {
        const int row = tm + r + half * 8;
        const int col = tn + l;
        float val = acc[r];
        if (bias) val += bias[col];
        Cp[(long long)row * ldc + col] = val;
    }
}

// Fused disentangled attention. One wave per 16-row i-tile of one (b,h).
// grid = (S/16/4, H, B), block = 128 (4 waves).
__global__ void deberta_attn(const float* __restrict__ q, const float* __restrict__ kmat,
                             const float* __restrict__ vmat,
                             const float* __restrict__ c2p, const float* __restrict__ p2c,
                             const long long* __restrict__ relpos,
                             const unsigned char* __restrict__ amask,
                             float* __restrict__ out)
{
    const float scale = 0.07216878364870323f; // 1/sqrt(3*D) = 1/sqrt(192)
    const float NEGINF = -__builtin_inff();

    const int wave = threadIdx.x >> 5;
    const int lane = threadIdx.x & 31;
    const int half = lane >> 4;
    const int l    = lane & 15;
    const int i0 = (blockIdx.x * 4 + wave) * 16;
    const int h  = blockIdx.y;
    const int b  = blockIdx.z;
    const long long bh = (long long)(b * H_ + h);

    __shared__ float pbuf[4][16 * 16];

    // Preload q fragments for this 16-row tile: 16 k-steps of (16x4).
    v2f qa[16];
    {
        const float* qrow = q + ((long long)(b * S_ + i0 + l)) * HID_ + h * D_;
#pragma unroll
        for (int kk = 0; kk < 16; ++kk) {
            qa[kk].x = qrow[kk * 4 + half * 2];
            qa[kk].y = qrow[kk * 4 + half * 2 + 1];
        }
    }

    v8f acc[4] = {v8f{}, v8f{}, v8f{}, v8f{}};
    float rm[8], rs[8];
#pragma unroll
    for (int r = 0; r < 8; ++r) { rm[r] = NEGINF; rs[r] = 0.0f; }

    for (int jt = 0; jt < S_ / 16; ++jt) {
        const int j0 = jt * 16;

        // ---- c2c = q_tile (16x64) @ k_tile^T (64x16) ----
        v8f c = {};
        {
            const float* krow = kmat + ((long long)(b * S_ + j0 + l)) * HID_ + h * D_;
#pragma unroll
            for (int kk = 0; kk < 16; ++kk) {
                v2f bb; bb.x = krow[kk * 4 + half * 2]; bb.y = krow[kk * 4 + half * 2 + 1];
                c = __builtin_amdgcn_wmma_f32_16x16x4_f32(false, qa[kk], false, bb,
                                                          (short)0, c, false, false);
            }
        }

        // ---- add gathered c2p/p2c, scale, mask; online softmax update ----
        float pr[8];
#pragma unroll
        for (int r = 0; r < 8; ++r) {
            const int row = i0 + r + half * 8;
            const int col = j0 + l;
            const int rp = iclamp((int)relpos[(long long)row * S_ + col], 0, TK_ - 1);
            float sval;
            if (amask[(long long)b * S_ * S_ + (long long)row * S_ + col]) {
                const float c2pv = c2p[(bh * S_ + row) * TK_ + rp];
                const float p2cv = p2c[(bh * S_ + col) * TK_ + rp];
                sval = (c[r] + c2pv + p2cv) * scale;
            } else {
                sval = NEGINF;
            }
            pr[r] = sval;
        }
#pragma unroll
        for (int r = 0; r < 8; ++r) {
            float m = pr[r];
#pragma unroll
            for (int off = 1; off < 16; off <<= 1)
                m = fmaxf(m, __shfl_xor(m, off, 32));
            const float mnew  = fmaxf(rm[r], m);
            const float alpha = (rm[r] == NEGINF) ? 0.0f : __expf(rm[r] - mnew);
            float pv = (pr[r] == NEGINF) ? 0.0f : __expf(pr[r] - mnew);
            float s = pv;
#pragma unroll
            for (int off = 1; off < 16; off <<= 1)
                s += __shfl_xor(s, off, 32);
            rs[r] = rs[r] * alpha + s;
            rm[r] = mnew;
            pr[r] = pv;
#pragma unroll
            for (int nt = 0; nt < 4; ++nt) acc[nt][r] *= alpha;
        }

        // ---- relayout probs C-layout -> A-layout via LDS ----
#pragma unroll
        for (int r = 0; r < 8; ++r)
            pbuf[wave][(r + half * 8) * 16 + l] = pr[r];
        __syncthreads();

        v2f pa[4];
#pragma unroll
        for (int kk = 0; kk < 4; ++kk) {
            pa[kk].x = pbuf[wave][l * 16 + kk * 4 + half * 2];
            pa[kk].y = pbuf[wave][l * 16 + kk * 4 + half * 2 + 1];
        }

        // ---- acc += probs (16x16) @ v_tile (16x64) ----
#pragma unroll
        for (int nt = 0; nt < 4; ++nt) {
#pragma unroll
            for (int kk = 0; kk < 4; ++kk) {
                const int jrow = j0 + kk * 4 + half * 2;
                const float* vr = vmat + ((long long)(b * S_ + jrow)) * HID_ + h * D_ + nt * 16 + l;
                v2f bb; bb.x = vr[0]; bb.y = vr[HID_];
                acc[nt] = __builtin_amdgcn_wmma_f32_16x16x4_f32(false, pa[kk], false, bb,
                                                                (short)0, acc[nt], false, false);
            }
        }
        __syncthreads();
    }

    // ---- normalize and store (B,S,HID) ----
#pragma unroll
    for (int r = 0; r < 8; ++r) {
        const int row = i0 + r + half * 8;
        const float inv = (rs[r] > 0.0f) ? 1.0f / rs[r] : 0.0f;
#pragma unroll
        for (int nt = 0; nt < 4; ++nt)
            out[((long long)(b * S_ + row)) * HID_ + h * D_ + nt * 16 + l] = acc[nt][r] * inv;
    }
}

extern "C" void kernel_launch(void* const* d_in, const int* in_sizes, int n_in,
                              void* d_out, int out_size, void* d_ws, size_t ws_size,
                              hipStream_t stream) {
    (void)in_sizes; (void)n_in; (void)out_size; (void)ws_size;

    const float*         hs     = (const float*)d_in[0];
    const unsigned char* amask  = (const unsigned char*)d_in[1]; // bool (B,1,S,S)
    const long long*     relpos = (const long long*)d_in[2];     // int64 (S,S)
    const float*         relemb = (const float*)d_in[3];         // (2K, HID)
    const float*         Wq     = (const float*)d_in[4];
    const float*         bq     = (const float*)d_in[5];
    const float*         Wk     = (const float*)d_in[6];
    const float*         Wv     = (const float*)d_in[7];
    const float*         bv     = (const float*)d_in[8];
    const float*         Wpos   = (const float*)d_in[9];
    const float*         Wposq  = (const float*)d_in[10];
    const float*         bposq  = (const float*)d_in[11];

    float* ws = (float*)d_ws;
    const long long nQ   = (long long)B_ * S_ * HID_;      // 3,145,728
    const long long nPos = (long long)TK_ * HID_;          //   393,216
    const long long nP   = (long long)B_ * H_ * S_ * TK_;  // 25,165,824
    float* qbuf    = ws;
    float* kbuf    = qbuf + nQ;
    float* vbuf    = kbuf + nQ;
    float* posqbuf = vbuf + nQ;
    float* poskbuf = posqbuf + nPos;
    float* c2pbuf  = poskbuf + nPos;
    float* p2cbuf  = c2pbuf + nP;

    const dim3 blk(128);

    // ---- projections: Y = X @ W^T (+ bias) ----
    // hidden (4096 x 768) x (768 x 768)^T
    wmma_gemm_abt<<<dim3(HID_ / 64, (B_ * S_) / 16, 1), blk, 0, stream>>>(
        hs, Wq, bq, qbuf, B_ * S_, HID_, HID_, HID_, HID_, HID_, 1, 0, 0, 0, 0);
    wmma_gemm_abt<<<dim3(HID_ / 64, (B_ * S_) / 16, 1), blk, 0, stream>>>(
        hs, Wk, nullptr, kbuf, B_ * S_, HID_, HID_, HID_, HID_, HID_, 1, 0, 0, 0, 0);
    wmma_gemm_abt<<<dim3(HID_ / 64, (B_ * S_) / 16, 1), blk, 0, stream>>>(
        hs, Wv, bv, vbuf, B_ * S_, HID_, HID_, HID_, HID_, HID_, 1, 0, 0, 0, 0);
    // rel_embeddings (512 x 768) projections
    wmma_gemm_abt<<<dim3(HID_ / 64, TK_ / 16, 1), blk, 0, stream>>>(
        relemb, Wposq, bposq, posqbuf, TK_, HID_, HID_, HID_, HID_, HID_, 1, 0, 0, 0, 0);
    wmma_gemm_abt<<<dim3(HID_ / 64, TK_ / 16, 1), blk, 0, stream>>>(
        relemb, Wpos, nullptr, poskbuf, TK_, HID_, HID_, HID_, HID_, HID_, 1, 0, 0, 0, 0);

    // ---- positional score tables per (b,h): (S x 2K), K = D = 64 ----
    // c2p_full[b,h,i,t] = q[b,i,hD:] . pos_k[t,hD:]
    wmma_gemm_abt<<<dim3(TK_ / 64, S_ / 16, B_ * H_), blk, 0, stream>>>(
        qbuf, poskbuf, nullptr, c2pbuf, S_, TK_, D_, HID_, HID_, TK_,
        H_, (long long)S_ * HID_, (long long)D_, (long long)D_, (long long)S_ * TK_);
    // p2c_full[b,h,j,t] = k[b,j,hD:] . pos_q[t,hD:]
    wmma_gemm_abt<<<dim3(TK_ / 64, S_ / 16, B_ * H_), blk, 0, stream>>>(
        kbuf, posqbuf, nullptr, p2cbuf, S_, TK_, D_, HID_, HID_, TK_,
        H_, (long long)S_ * HID_, (long long)D_, (long long)D_, (long long)S_ * TK_);

    // ---- fused attention (c2c + gathers + softmax + PV) ----
    deberta_attn<<<dim3(S_ / 16 / 4, H_, B_), blk, 0, stream>>>(
        qbuf, kbuf, vbuf, c2pbuf, p2cbuf, relpos, amask, (float*)d_out);
}